// GMP_61967788146742
// MI455X (gfx1250) — compile-verified
//
#include <hip/hip_runtime.h>
#include <math.h>

typedef float v2f __attribute__((ext_vector_type(2)));
typedef float v8f __attribute__((ext_vector_type(8)));

#define Bn     8
#define Nn     32768
#define Cn     4
#define Mn     11
#define Dn     5
#define Kn     55      // D*M
#define KPAD   56      // padded to multiple of 4
#define NCHUNK 14      // KPAD / 4
#define TILE_N 256     // outputs per workgroup
#define HALO   (Mn - 1)            // 10 history samples
#define PITCH  272                 // LDS row pitch (v2f units), padded vs 266

// Feature-plane element offset (v2f units) for flattened basis index k:
// k = i*11 + j  ->  plane i, tap shift j.  Compile-time when k is constant.
#define FOFF(k) ((((k) / 11) * PITCH) + ((k) % 11))

static __device__ inline v2f mk2(float a, float b) { v2f r; r.x = a; r.y = b; return r; }

__global__ __launch_bounds__(256)
void gmp_wmma_kernel(const float* __restrict__ x,      // [B,N,C,2]
                     const float* __restrict__ weight, // [C,K]
                     float* __restrict__ out) {        // [B,N,C,2]
    // G[i][t] = (re*|s|^i, im*|s|^i) for sample t (t=0 -> global n0-10).
    // Plane 5 is a zero plane backing the k=55 pad reads (W[55]=0 kills the
    // contribution, but the value must be finite, not garbage).
    __shared__ v2f   G[6 * PITCH];
    __shared__ float Wl[64];

    const int tid  = threadIdx.x;
    const int tpb  = Nn / TILE_N;              // tiles per (b,c)
    const int tile = blockIdx.x % tpb;
    const int bc   = blockIdx.x / tpb;
    const int c    = bc % Cn;
    const int b    = bc / Cn;
    const int n0   = tile * TILE_N;

    // ---------- Phase 1: stage features + weights into LDS ----------
    for (int t = tid; t < TILE_N + HALO; t += 256) {
        const int gt = n0 + t - HALO;          // global sample index
        float re = 0.0f, im = 0.0f;
        if (gt >= 0) {
            const float* p = x + (((size_t)b * Nn + gt) * Cn + c) * 2;
            re = p[0];
            im = p[1];
        }
        const float a  = sqrtf(re * re + im * im);
        const float a2 = a * a;
        const float a3 = a2 * a;
        const float a4 = a2 * a2;
        G[0 * PITCH + t] = mk2(re,      im);
        G[1 * PITCH + t] = mk2(re * a,  im * a);
        G[2 * PITCH + t] = mk2(re * a2, im * a2);
        G[3 * PITCH + t] = mk2(re * a3, im * a3);
        G[4 * PITCH + t] = mk2(re * a4, im * a4);
    }
    for (int t = tid; t < PITCH; t += 256) G[5 * PITCH + t] = mk2(0.0f, 0.0f);
    if (tid < 64) Wl[tid] = (tid < Kn) ? weight[c * Kn + tid] : 0.0f;
    __syncthreads();

    // ---------- Phase 2: WMMA matvec chains ----------
    const int lane  = tid & 31;         // wave32
    const int wv    = tid >> 5;         // 8 waves per workgroup
    const int lcol  = lane & 15;        // output column (n within 16-tile)
    const bool hi   = (lane >> 4) != 0; // selects K pair {2,3} vs {0,1}
    const int tb0   = wv * 16 + lcol;   // feature-sample base, it==0

    // Fully resolved per-lane LDS pointers for B fragments (it==0); the
    // second output iteration is a constant +128 v2f (= +1024B) which folds
    // into the ds_load immediate offset.  A fragments (weights, broadcast
    // over rows) are register-resident.
    const v2f* gpx[NCHUNK];
    const v2f* gpy[NCHUNK];
    v2f wfrag[NCHUNK];
#pragma unroll
    for (int chunk = 0; chunk < NCHUNK; ++chunk) {
        const int k0 = chunk * 4;
        gpx[chunk] = G + (hi ? FOFF(k0 + 2) : FOFF(k0 + 0)) + tb0;
        gpy[chunk] = G + (hi ? FOFF(k0 + 3) : FOFF(k0 + 1)) + tb0;
        const int kb = k0 + (hi ? 2 : 0);
        wfrag[chunk] = mk2(Wl[kb], Wl[kb + 1]);
    }

#pragma unroll
    for (int it = 0; it < 2; ++it) {
        v8f acc_re = {};
        v8f acc_im = {};

#pragma unroll
        for (int chunk = 0; chunk < NCHUNK; ++chunk) {
            const v2f g1 = gpx[chunk][it * 128];   // immediate offset 0 / 1024B
            const v2f g2 = gpy[chunk][it * 128];
            const v2f bre = mk2(g1.x, g2.x);
            const v2f bim = mk2(g1.y, g2.y);

            acc_re = __builtin_amdgcn_wmma_f32_16x16x4_f32(
                false, wfrag[chunk], false, bre, (short)0, acc_re, false, false);
            acc_im = __builtin_amdgcn_wmma_f32_16x16x4_f32(
                false, wfrag[chunk], false, bim, (short)0, acc_im, false, false);
        }

        // D[0, col] sits in accumulator VGPR0 of lane==col (lanes 0-15).
        if (lane < 16) {
            const int n = n0 + it * 128 + wv * 16 + lcol;
            float* po = out + (((size_t)b * Nn + n) * Cn + c) * 2;
            po[0] = acc_re[0];
            po[1] = acc_im[0];
        }
    }
}

extern "C" void kernel_launch(void* const* d_in, const int* in_sizes, int n_in,
                              void* d_out, int out_size, void* d_ws, size_t ws_size,
                              hipStream_t stream) {
    (void)in_sizes; (void)n_in; (void)out_size; (void)d_ws; (void)ws_size;
    const float* x = (const float*)d_in[0];
    const float* w = (const float*)d_in[1];
    float* out     = (float*)d_out;

    const int blocks = Bn * Cn * (Nn / TILE_N);   // 4096 workgroups
    gmp_wmma_kernel<<<dim3(blocks), dim3(256), 0, stream>>>(x, w, out);
}